// HoloLinear_v3_31404800868920
// MI455X (gfx1250) — compile-verified
//
#include <hip/hip_runtime.h>
#include <hip/hip_bf16.h>

typedef __attribute__((ext_vector_type(2))) float v2f;
typedef __attribute__((ext_vector_type(8))) float v8f;

#define B_    2
#define S_    128
#define IN_   512
#define OUT_  1024
#define BS_   (B_ * S_)
#define TOPK_ 4

// ---------------------------------------------------------------------------
// Kernel 1: out_phi[bs, o, t] = sum_k W[o,k] * phi[bs, k, t]
// Per-wave: 64(o) x 64(t) output block = 4x4 tiles of V_WMMA_F32_16X16X4_F32.
// A (16x4 f32): lanes 0-15 hold M=0..15 with VGPR{0,1}=K{0,1}; lanes 16-31
// hold K{2,3}  -> contiguous float2 (b64) load from a W row.
// B (4x16 f32): lanes 0-15 hold N with VGPR{0,1}=K{0,1}; lanes 16-31 K{2,3}
// -> two b32 loads from phi (row stride S_).
// 16 WMMAs per k-step vs 12 VMEM ops; 128 acc VGPRs, no spill.
// ---------------------------------------------------------------------------
__global__ void __launch_bounds__(256)
holo_phi_wmma_kernel(const float* __restrict__ phi,
                     const float* __restrict__ W,
                     float* __restrict__ out_phi)
{
    const int lane = threadIdx.x & 31;
    const int wid  = blockIdx.x * 8 + (threadIdx.x >> 5);   // global wave id
    const int bs   = wid >> 5;                              // 32 wave-tasks per bs
    const int rem  = wid & 31;
    const int oBase = (rem >> 1) << 6;                      // 16 o-blocks * 64
    const int tBase = (rem & 1)  << 6;                      // 2  t-blocks * 64
    const int l16  = lane & 15;
    const int hsel = lane >> 4;                             // 0: K0/1, 1: K2/3

    const float* __restrict__ phi_bs = phi + (size_t)bs * (IN_ * S_);

    const float* Arow[4];
#pragma unroll
    for (int r = 0; r < 4; ++r)
        Arow[r] = W + (size_t)(oBase + r * 16 + l16) * IN_ + 2 * hsel;

    const float* Bcol[4];
#pragma unroll
    for (int c = 0; c < 4; ++c)
        Bcol[c] = phi_bs + (size_t)(2 * hsel) * S_ + (tBase + c * 16 + l16);

    v8f acc[4][4];
#pragma unroll
    for (int r = 0; r < 4; ++r)
#pragma unroll
        for (int c = 0; c < 4; ++c)
            acc[r][c] = (v8f){};

#pragma unroll 2
    for (int k = 0; k < IN_; k += 4) {
        v2f a[4];
#pragma unroll
        for (int r = 0; r < 4; ++r)
            a[r] = *(const v2f*)(Arow[r] + k);

        const int kb = k * S_;
        v2f b[4];
#pragma unroll
        for (int c = 0; c < 4; ++c) {
            b[c].x = Bcol[c][kb];
            b[c].y = Bcol[c][kb + S_];
        }

#pragma unroll
        for (int r = 0; r < 4; ++r)
#pragma unroll
            for (int c = 0; c < 4; ++c)
                acc[r][c] = __builtin_amdgcn_wmma_f32_16x16x4_f32(
                    false, a[r], false, b[c], (short)0, acc[r][c], false, false);
    }

    // C/D layout: VGPR v -> lanes 0-15: (M=v, N=l16); lanes 16-31: (M=v+8, N=l16)
    float* __restrict__ outp = out_phi + (size_t)bs * (OUT_ * S_);
#pragma unroll
    for (int r = 0; r < 4; ++r)
#pragma unroll
        for (int c = 0; c < 4; ++c)
#pragma unroll
            for (int v = 0; v < 8; ++v) {
                const int o = oBase + r * 16 + v + 8 * hsel;
                const int t = tBase + c * 16 + l16;
                outp[(size_t)o * S_ + t] = acc[r][c][v];
            }
}

// ---------------------------------------------------------------------------
// Kernel 2: per (bs,o):
//   out_signal = sum_i signal[bs,i]*W[o,i]   (summands reused for top-k)
//   out_beta   = sum_i beta[bs,i]*W[o,i]
//   out_pi     = top-4 indices of |signal[bs,i]*W[o,i]| (strict >, stable)
// ---------------------------------------------------------------------------
__device__ __forceinline__ void topk_insert(float p, int idx,
                                            float& v0, float& v1, float& v2, float& v3,
                                            int& i0, int& i1, int& i2, int& i3)
{
    if (p > v3) {
        if (p > v2) {
            v3 = v2; i3 = i2;
            if (p > v1) {
                v2 = v1; i2 = i1;
                if (p > v0) { v1 = v0; i1 = i0; v0 = p; i0 = idx; }
                else        { v1 = p;  i1 = idx; }
            } else { v2 = p; i2 = idx; }
        } else { v3 = p; i3 = idx; }
    }
}

__global__ void __launch_bounds__(256)
holo_topk_kernel(const float* __restrict__ signal,
                 const float* __restrict__ beta,
                 const float* __restrict__ W,
                 float* __restrict__ out_signal,
                 float* __restrict__ out_beta,
                 short* __restrict__ out_pi)
{
    const int bs = blockIdx.x >> 2;                         // 0..255
    const int o  = ((blockIdx.x & 3) << 8) + threadIdx.x;   // 0..1023

    const float4* __restrict__ s4 = (const float4*)(signal + (size_t)bs * IN_);
    const float4* __restrict__ b4 = (const float4*)(beta   + (size_t)bs * IN_);
    const float4* __restrict__ w4 = (const float4*)(W      + (size_t)o  * IN_);

    float sum = 0.f, sumb = 0.f;
    float v0 = -1.f, v1 = -1.f, v2 = -1.f, v3 = -1.f;       // |p| >= 0 always wins
    int   i0 = 0, i1 = 1, i2 = 2, i3 = 3;

#pragma unroll 4
    for (int q = 0; q < IN_ / 4; ++q) {
        const float4 s = s4[q];
        const float4 b = b4[q];
        const float4 w = w4[q];
        const int base = q * 4;

        float p;
        p = s.x * w.x; sum += p; sumb += b.x * w.x;
        topk_insert(fabsf(p), base + 0, v0, v1, v2, v3, i0, i1, i2, i3);
        p = s.y * w.y; sum += p; sumb += b.y * w.y;
        topk_insert(fabsf(p), base + 1, v0, v1, v2, v3, i0, i1, i2, i3);
        p = s.z * w.z; sum += p; sumb += b.z * w.z;
        topk_insert(fabsf(p), base + 2, v0, v1, v2, v3, i0, i1, i2, i3);
        p = s.w * w.w; sum += p; sumb += b.w * w.w;
        topk_insert(fabsf(p), base + 3, v0, v1, v2, v3, i0, i1, i2, i3);
    }

    const size_t oo = (size_t)bs * OUT_ + o;
    out_signal[oo] = sum;
    out_beta[oo]   = sumb;
    short* pi = out_pi + oo * TOPK_;
    pi[0] = (short)i0; pi[1] = (short)i1; pi[2] = (short)i2; pi[3] = (short)i3;
}

extern "C" void kernel_launch(void* const* d_in, const int* in_sizes, int n_in,
                              void* d_out, int out_size, void* d_ws, size_t ws_size,
                              hipStream_t stream)
{
    (void)in_sizes; (void)n_in; (void)out_size; (void)d_ws; (void)ws_size;

    const float* signal = (const float*)d_in[0];   // [B,S,IN]
    const float* phi    = (const float*)d_in[1];   // [B,S,IN,S]
    const float* beta   = (const float*)d_in[2];   // [B,S,IN,1]
    const float* W      = (const float*)d_in[3];   // [OUT,IN]

    // Outputs concatenated flat in return order: signal, phi, beta, pi(int16)
    float* out    = (float*)d_out;
    float* o_sig  = out;                                        // BS*OUT
    float* o_phi  = o_sig  + (size_t)BS_ * OUT_;                // BS*OUT*S
    float* o_beta = o_phi  + (size_t)BS_ * OUT_ * S_;           // BS*OUT
    short* o_pi   = (short*)(o_beta + (size_t)BS_ * OUT_);      // BS*OUT*4 int16

    // 8192 wave-tasks (256 bs * 16 o-blocks * 2 t-blocks), 8 waves/block
    holo_phi_wmma_kernel<<<1024, 256, 0, stream>>>(phi, W, o_phi);

    // 256 bs * 1024 o threads
    holo_topk_kernel<<<1024, 256, 0, stream>>>(signal, beta, W, o_sig, o_beta, o_pi);
}